// DSTFT_7808250544780
// MI455X (gfx1250) — compile-verified
//
#include <hip/hip_runtime.h>
#include <math.h>

#define B_SZ 4
#define L_SZ 512000
#define N_SZ 512
#define T_SZ 3997
#define F_SZ 257
#define WIN_MIN 25.6f
#define WIN_MAX 512.0f
#define TWO_PI 6.28318530717958647692f

typedef __attribute__((ext_vector_type(2))) float v2f;
typedef __attribute__((ext_vector_type(8))) float v8f;

// ---------------- per-frame start index + fractional offset ----------------
__global__ void k_prep(const float* __restrict__ strides,
                       int* __restrict__ start, float* __restrict__ frac) {
    int t = blockIdx.x * blockDim.x + threadIdx.x;
    if (t >= T_SZ) return;
    float s = fminf(fmaxf(strides[0], 0.0f), 512.0f);
    float frame = (t == 0) ? 0.0f : (float)t * s;
    float fl = floorf(frame);
    start[t] = (int)fl;
    frac[t] = frame - fl;
}

// ---------------- DFT cos/sin tables, exact mod-N phase reduction ----------
__global__ void k_tables(float* __restrict__ cosT, float* __restrict__ sinT) {
    int id = blockIdx.x * blockDim.x + threadIdx.x;
    if (id >= N_SZ * N_SZ) return;
    int f = id >> 9;
    int n = id & (N_SZ - 1);
    int m = (f * n) & (N_SZ - 1);             // exact 2*pi reduction
    float a = (float)m * (TWO_PI / (float)N_SZ);
    float s, c;
    sincosf(a, &s, &c);
    cosT[id] = c;
    sinT[id] = s;
}

// ---------------- normalized, shifted, powered Hann taper (N x T) ----------
__global__ __launch_bounds__(256) void k_tap(const float* __restrict__ frac,
                                             const float* __restrict__ win_length,
                                             const float* __restrict__ win_pow,
                                             float* __restrict__ tap) {
    __shared__ float red[256];
    int t = blockIdx.x;
    float fr = frac[t];
    float wl = fminf(fmaxf(win_length[0], WIN_MIN), WIN_MAX);
    float p  = win_pow[0];
    float hi = ceilf((511.0f + wl) / 2.0f);
    float lo = floorf((511.0f - wl) / 2.0f);
    float shift = (wl - 511.0f) / 2.0f;       // (wl - N + 1)/2

    float v[2];
    float lsum = 0.0f;
#pragma unroll
    for (int j = 0; j < 2; ++j) {
        int n = threadIdx.x + j * 256;
        float base = (float)n - fr;
        float tv = 0.5f - 0.5f * cosf((TWO_PI * (base + shift)) / wl);
        if (base >= hi || base <= lo) tv = 0.0f;
        v[j] = tv;
        lsum += tv;
    }
    red[threadIdx.x] = lsum;
    __syncthreads();
    for (int off = 128; off > 0; off >>= 1) {
        if (threadIdx.x < off) red[threadIdx.x] += red[threadIdx.x + off];
        __syncthreads();
    }
    float sum = red[0];
#pragma unroll
    for (int j = 0; j < 2; ++j) {
        int n = threadIdx.x + j * 256;
        float tv = v[j] / sum;
        if (p != 1.0f) tv = powf(tv, p);
        tap[t * N_SZ + n] = tv;               // [t][n], coalesced along n
    }
}

// ---------------- fused taper-gather + f32 WMMA DFT + epilogue -------------
// Block tile: 64 f x 64 t, 8 waves, each wave = 16f x 32t (2 WMMA tiles).
// LDS rows use an EVEN stride (KC+2) and B is stored t-major, so every
// (k,k+1) fragment pair is one 8B-aligned ds_load_b64 straight into the
// even-aligned VGPR pair WMMA needs (no v_mov marshaling). Stride 34 floats
// -> lanes hit 16 distinct bank pairs: conflict-free.
#define KC 32
#define KP (KC + 2)
#define TM 64
#define TN 64

__global__ __launch_bounds__(256) void k_dft(const float* __restrict__ x,
                                             const int* __restrict__ start,
                                             const float* __restrict__ tap,
                                             const float* __restrict__ cosT,
                                             const float* __restrict__ sinT,
                                             float* __restrict__ out) {
    __shared__ float Ac[TM][KP];
    __shared__ float As[TM][KP];
    __shared__ float Bs[TN][KP];   // t-major

    const int tid   = threadIdx.x;
    const int lane  = tid & 31;
    const int wave  = tid >> 5;
    const int half  = lane >> 4;
    const int m16   = lane & 15;
    const int waveM = wave & 3;   // 4 strips along f
    const int waveN = wave >> 2;  // 2 strips along t (each strip = 2 tiles)

    const int t0 = blockIdx.x * TN;
    const int f0 = blockIdx.y * TM;
    const int b  = blockIdx.z;

    v8f accR0 = {}, accR1 = {};
    v8f accI0 = {}, accI1 = {};

    const int colA = tid & 31;    // k within chunk
    const int rowA = tid >> 5;    // f row base (step 8)
    const int nB   = tid & 31;    // k within chunk
    const int tB   = tid >> 5;    // t row base (step 8)

    for (int k0 = 0; k0 < N_SZ; k0 += KC) {
        // stage cos/sin A-tiles (coalesced along k)
#pragma unroll
        for (int j = 0; j < 8; ++j) {
            int r  = rowA + j * 8;
            int gi = (f0 + r) * N_SZ + k0 + colA;
            Ac[r][colA] = cosT[gi];
            As[r][colA] = sinT[gi];
        }
        // stage tapered B-tile: x[start[t]+n] * tap[t][n] (coalesced along n)
#pragma unroll
        for (int j = 0; j < 8; ++j) {
            int tl = tB + j * 8;
            int t  = t0 + tl;
            float v = 0.0f;
            if (t < T_SZ) {
                int idx = start[t] + k0 + nB;
                if (idx < L_SZ)
                    v = x[b * L_SZ + idx] * tap[t * N_SZ + k0 + nB];
            }
            Bs[tl][nB] = v;
        }
        __syncthreads();

        {
            const int fr  = waveM * 16 + m16;
            const int tc0 = waveN * 32 + m16;
            const int tc1 = tc0 + 16;
#pragma unroll
            for (int kk = 0; kk < KC; kk += 4) {
                int ka = kk + 2 * half;   // ISA 16x4 f32 A layout: V0=K0/K2, V1=K1/K3
                v2f a_c = *(const v2f*)&Ac[fr][ka];
                v2f a_s = *(const v2f*)&As[fr][ka];
                v2f b0  = *(const v2f*)&Bs[tc0][ka];
                v2f b1  = *(const v2f*)&Bs[tc1][ka];
                accR0 = __builtin_amdgcn_wmma_f32_16x16x4_f32(false, a_c, false, b0,
                                                              (short)0, accR0, false, false);
                accR1 = __builtin_amdgcn_wmma_f32_16x16x4_f32(false, a_c, false, b1,
                                                              (short)0, accR1, false, false);
                accI0 = __builtin_amdgcn_wmma_f32_16x16x4_f32(false, a_s, false, b0,
                                                              (short)0, accI0, false, false);
                accI1 = __builtin_amdgcn_wmma_f32_16x16x4_f32(false, a_s, false, b1,
                                                              (short)0, accI1, false, false);
            }
        }
        __syncthreads();
    }

    // output offsets (floats) in return order: spec, stft(complex), real, imag, phase
    const size_t OFF_STFT  = (size_t)B_SZ * F_SZ * T_SZ;
    const size_t OFF_REAL  = OFF_STFT + (size_t)2 * B_SZ * N_SZ * T_SZ;
    const size_t OFF_IMAG  = OFF_REAL + (size_t)B_SZ * N_SZ * T_SZ;
    const size_t OFF_PHASE = OFF_IMAG + (size_t)B_SZ * N_SZ * T_SZ;

#pragma unroll
    for (int q = 0; q < 2; ++q) {
        int t = t0 + waveN * 32 + q * 16 + m16;
        if (t < T_SZ) {
#pragma unroll
            for (int i = 0; i < 8; ++i) {
                int f = f0 + waveM * 16 + half * 8 + i;  // C/D: V_i -> M=i (+8 hi half)
                float r  = q ? accR1[i] : accR0[i];
                float im = -(q ? accI1[i] : accI0[i]);   // imag = -sum(tapered*sin)
                size_t base = ((size_t)(b * N_SZ + f)) * T_SZ + t;
                out[OFF_REAL + base] = r;
                out[OFF_IMAG + base] = im;
                out[OFF_STFT + 2 * base]     = r;
                out[OFF_STFT + 2 * base + 1] = im;
                if (f < F_SZ) {
                    size_t sb = ((size_t)(b * F_SZ + f)) * T_SZ + t;
                    out[sb]             = sqrtf(r * r + im * im) + 1.1920929e-7f;
                    out[OFF_PHASE + sb] = atan2f(im, r);
                }
            }
        }
    }
}

extern "C" void kernel_launch(void* const* d_in, const int* in_sizes, int n_in,
                              void* d_out, int out_size, void* d_ws, size_t ws_size,
                              hipStream_t stream) {
    const float* x          = (const float*)d_in[0];
    const float* strides    = (const float*)d_in[1];
    const float* win_length = (const float*)d_in[2];
    const float* win_pow    = (const float*)d_in[3];
    float* out = (float*)d_out;

    char* ws = (char*)d_ws;
    int*   start = (int*)ws;                                   // T ints
    float* frac  = (float*)(ws + 16384);                       // T floats
    float* tap   = (float*)(ws + 32768);                       // T*N floats
    float* cosT  = (float*)(ws + 32768 + (size_t)T_SZ * N_SZ * 4);
    float* sinT  = cosT + (size_t)N_SZ * N_SZ;

    k_prep<<<(T_SZ + 255) / 256, 256, 0, stream>>>(strides, start, frac);
    k_tables<<<(N_SZ * N_SZ) / 256, 256, 0, stream>>>(cosT, sinT);
    k_tap<<<T_SZ, 256, 0, stream>>>(frac, win_length, win_pow, tap);

    dim3 grid((T_SZ + TN - 1) / TN, N_SZ / TM, B_SZ);
    k_dft<<<grid, 256, 0, stream>>>(x, start, tap, cosT, sinT, out);
}